// GlobalPointer_84241488543818
// MI455X (gfx1250) — compile-verified
//
#include <hip/hip_runtime.h>

// GlobalPointer head for MI455X (gfx1250, wave32, WMMA), 2x2 register-blocked.
//   K1: out = hs @ W + b  (8192x768x1152 fp32 GEMM, v_wmma_f32_16x16x4_f32)
//       fused interleaved RoPE, scatter Q -> ws[b][t][s][d], K^T -> ws[b][t][d][s]
//   K2: logits[b,t,m,n] = (Q_bt K_bt^T)/8 - masks, NT stores (144 x 512x512x64)

#define TT 9
#define HEAD 64
#define BB 16
#define SS 512
#define HH 768
#define NPROJ (TT * HEAD * 2) /* 1152 */
#define NEGC 1000000000000.0f
#define LOG1E4_OVER_32 0.28782313662425574f /* ln(10000)/32 */

typedef float v2f __attribute__((ext_vector_type(2)));
typedef float v8f __attribute__((ext_vector_type(8)));

#define WMMA_F32(A, B, C) \
  __builtin_amdgcn_wmma_f32_16x16x4_f32(false, (A), false, (B), (short)0, (C), false, false)

// Epilogue for one 16x16 tile of the projection: bias + interleaved RoPE +
// scatter. Column decomposition: nc = t*128 + d*2 + qk. RoPE partner element
// (col nc^2) lives in lane^2 of the same wave half.
__device__ __forceinline__ void rope_scatter(
    const v8f& acc, int mbase, int nc, int half,
    const float* __restrict__ bias,
    float* __restrict__ qbuf, float* __restrict__ kbuf)
{
  const int t  = nc >> 7;
  const int d  = (nc & 127) >> 1;
  const int qk = nc & 1;
  const int f  = (nc & 127) >> 2;                   // frequency index d/2
  const float invfreq = __expf(-(float)f * LOG1E4_OVER_32);
  const float bval = bias[nc];
#pragma unroll
  for (int i = 0; i < 8; ++i) {
    float x  = acc[i] + bval;
    float xp = __shfl_xor(x, 2, 32);
    const int row = mbase + i + 8 * half;           // global row in [0, B*S)
    const int s   = row & (SS - 1);
    const int b   = row >> 9;
    float sn, cs;
    __sincosf((float)s * invfreq, &sn, &cs);
    const float val = (nc & 2) ? (x * cs + xp * sn) : (x * cs - xp * sn);
    if (qk) kbuf[(((size_t)b * TT + t) * HEAD + d) * SS + s] = val;  // K^T
    else    qbuf[(((size_t)b * TT + t) * SS + s) * HEAD + d] = val;  // Q
  }
}

// ---------------------------------------------------------------------------
// Kernel 1: projection GEMM + bias + RoPE. Block = 128 thr = 4 waves in 2x2;
// each wave computes a 32x32 tile (4 accumulators, 2x2 fragment reuse).
// Grid = (NPROJ/64, B*S/64) = (18, 128).
// ---------------------------------------------------------------------------
__global__ __launch_bounds__(128) void gp_proj_rope(
    const float* __restrict__ hs,    // [B*S, H]
    const float* __restrict__ W,     // [H, NPROJ]
    const float* __restrict__ bias,  // [NPROJ]
    float* __restrict__ qbuf,        // [B, T, S, D]
    float* __restrict__ kbuf)        // [B, T, D, S]
{
  const int lane = threadIdx.x & 31;
  const int wave = threadIdx.x >> 5;
  const int wy = wave >> 1, wx = wave & 1;
  const int half = lane >> 4;
  const int l15  = lane & 15;
  const int n0 = blockIdx.x * 64 + wx * 32;
  const int m0 = blockIdx.y * 64 + wy * 32;
  const int nc0 = n0 + l15, nc1 = n0 + 16 + l15;
  const int ar0 = m0 + l15, ar1 = m0 + 16 + l15;

  v8f acc00 = {}, acc01 = {}, acc10 = {}, acc11 = {};
  const float* a0p = hs + (size_t)ar0 * HH;
  const float* a1p = hs + (size_t)ar1 * HH;

#pragma unroll 2
  for (int k = 0; k < HH; k += 4) {
    const int kk = k + 2 * half;
    // A 16x4 frags: lanes 0-15 -> K=k,k+1 ; lanes 16-31 -> K=k+2,k+3
    v2f a0 = *(const v2f*)(a0p + kk);
    v2f a1 = *(const v2f*)(a1p + kk);
    const float* w0 = W + (size_t)kk * NPROJ;
    const float* w1 = W + (size_t)(kk + 1) * NPROJ;
    v2f b0; b0.x = w0[nc0]; b0.y = w1[nc0];
    v2f b1; b1.x = w0[nc1]; b1.y = w1[nc1];
    acc00 = WMMA_F32(a0, b0, acc00);
    acc01 = WMMA_F32(a0, b1, acc01);
    acc10 = WMMA_F32(a1, b0, acc10);
    acc11 = WMMA_F32(a1, b1, acc11);
  }

  rope_scatter(acc00, m0,      nc0, half, bias, qbuf, kbuf);
  rope_scatter(acc01, m0,      nc1, half, bias, qbuf, kbuf);
  rope_scatter(acc10, m0 + 16, nc0, half, bias, qbuf, kbuf);
  rope_scatter(acc11, m0 + 16, nc1, half, bias, qbuf, kbuf);
}

// Epilogue for one 16x16 logits tile: scale + pair mask + causal mask,
// non-temporal stores (streaming 151 MB output must not evict Q/K from L2).
__device__ __forceinline__ void logits_store(
    const v8f& acc, int mbase, int ncol, int half, int b,
    const float* __restrict__ amask, float* __restrict__ obase)
{
  const float am_n = amask[(size_t)b * SS + ncol];
#pragma unroll
  for (int i = 0; i < 8; ++i) {
    const int m = mbase + i + 8 * half;
    const float am_m = amask[(size_t)b * SS + m];
    float val = acc[i] * 0.125f;                    // 1/sqrt(64)
    val -= (1.0f - am_m * am_n) * NEGC;             // pair mask
    if (ncol < m) val -= NEGC;                      // keep upper triangle
    __builtin_nontemporal_store(val, &obase[(size_t)m * SS + ncol]);
  }
}

// ---------------------------------------------------------------------------
// Kernel 2: per-(b,t) GEMM logits = Q(512x64) * K^T(64x512)/8 + masks.
// Block = 128 thr = 4 waves in 2x2; each wave a 32x32 tile (4 accumulators).
// Grid = (512/64, 512/64, B*T) = (8, 8, 144).
// ---------------------------------------------------------------------------
__global__ __launch_bounds__(128) void gp_attn_logits(
    const float* __restrict__ qbuf,   // [B*T, S, D]
    const float* __restrict__ kbuf,   // [B*T, D, S]
    const float* __restrict__ amask,  // [B, S]
    float* __restrict__ out)          // [B, T, S, S]
{
  const int lane = threadIdx.x & 31;
  const int wave = threadIdx.x >> 5;
  const int wy = wave >> 1, wx = wave & 1;
  const int half = lane >> 4;
  const int l15  = lane & 15;
  const int n0 = blockIdx.x * 64 + wx * 32;
  const int m0 = blockIdx.y * 64 + wy * 32;
  const int bt = blockIdx.z;            // b*T + t
  const int b  = bt / TT;
  const int nc0 = n0 + l15, nc1 = n0 + 16 + l15;

  const float* q  = qbuf + (size_t)bt * SS * HEAD;
  const float* kt = kbuf + (size_t)bt * HEAD * SS;
  const float* q0 = q + (size_t)(m0 + l15) * HEAD;
  const float* q1 = q + (size_t)(m0 + 16 + l15) * HEAD;

  v8f acc00 = {}, acc01 = {}, acc10 = {}, acc11 = {};

#pragma unroll
  for (int k = 0; k < HEAD; k += 4) {
    const int kk = k + 2 * half;
    v2f a0 = *(const v2f*)(q0 + kk);
    v2f a1 = *(const v2f*)(q1 + kk);
    const float* k0 = kt + (size_t)kk * SS;
    const float* k1 = kt + (size_t)(kk + 1) * SS;
    v2f b0; b0.x = k0[nc0]; b0.y = k1[nc0];
    v2f b1; b1.x = k0[nc1]; b1.y = k1[nc1];
    acc00 = WMMA_F32(a0, b0, acc00);
    acc01 = WMMA_F32(a0, b1, acc01);
    acc10 = WMMA_F32(a1, b0, acc10);
    acc11 = WMMA_F32(a1, b1, acc11);
  }

  float* obase = out + (size_t)bt * SS * SS;
  logits_store(acc00, m0,      nc0, half, b, amask, obase);
  logits_store(acc01, m0,      nc1, half, b, amask, obase);
  logits_store(acc10, m0 + 16, nc0, half, b, amask, obase);
  logits_store(acc11, m0 + 16, nc1, half, b, amask, obase);
}

extern "C" void kernel_launch(void* const* d_in, const int* in_sizes, int n_in,
                              void* d_out, int out_size, void* d_ws, size_t ws_size,
                              hipStream_t stream) {
  const float* hs    = (const float*)d_in[0];  // [16,512,768]
  const float* amask = (const float*)d_in[1];  // [16,512]
  const float* W     = (const float*)d_in[2];  // [768,1152]
  const float* bias  = (const float*)d_in[3];  // [1152]
  float* out = (float*)d_out;                  // [16,9,512,512]

  float* qbuf = (float*)d_ws;                       // B*T*S*D floats
  float* kbuf = qbuf + (size_t)BB * TT * SS * HEAD; // B*T*D*S floats

  dim3 blk(128);
  dim3 g1(NPROJ / 64, (BB * SS) / 64);      // (18, 128)
  gp_proj_rope<<<g1, blk, 0, stream>>>(hs, W, bias, qbuf, kbuf);

  dim3 g2(SS / 64, SS / 64, BB * TT);       // (8, 8, 144)
  gp_attn_logits<<<g2, blk, 0, stream>>>(qbuf, kbuf, amask, out);
}